// SANLayer_24446953849543
// MI455X (gfx1250) — compile-verified
//
#include <hip/hip_runtime.h>
#include <hip/hip_bf16.h>

typedef __attribute__((ext_vector_type(2))) float v2f;
typedef __attribute__((ext_vector_type(8))) float v8f;

#define D 64            // D_IN == D_OUT == 64
#define NEG_SLOPE 0.2f

// ---------- monotonic float <-> uint encoding for exact atomic float max ----------
__device__ __forceinline__ unsigned fenc(float x) {
    unsigned u = __float_as_uint(x);
    return (u & 0x80000000u) ? ~u : (u | 0x80000000u);
}
__device__ __forceinline__ float fdec(unsigned e) {
    unsigned u = (e & 0x80000000u) ? (e ^ 0x80000000u) : ~e;
    return __uint_as_float(u);
}
#define ENC_NEG_INF 0x007FFFFFu   // fenc(-inf)

__device__ __forceinline__ float lrelu(float x) { return x > 0.0f ? x : NEG_SLOPE * x; }

// ---------------------------------------------------------------------------------
// init: zero d_out, zero denoms, set segment-max buffers to enc(-inf)
// ---------------------------------------------------------------------------------
__global__ void init_kernel(float* __restrict__ out, float* __restrict__ den_u,
                            float* __restrict__ den_d, unsigned* __restrict__ m_u,
                            unsigned* __restrict__ m_d, int n, int total_out) {
    int i = blockIdx.x * blockDim.x + threadIdx.x;
    if (i < total_out) out[i] = 0.0f;
    if (i < n) {
        den_u[i] = 0.0f;
        den_d[i] = 0.0f;
        m_u[i] = ENC_NEG_INF;
        m_d[i] = ENC_NEG_INF;
    }
}

// ---------------------------------------------------------------------------------
// Dense phase: h_gat = F@Wg + bg ; hp = F@Wp + bp ; s_src = h_gat@a_src ; s_dst = h_gat@a_dst
// One wave handles 16 node-rows. WMMA f32 16x16x4, K=64 -> 16 k-steps, 4 N-tiles,
// 2 weight matrices -> 128 v_wmma per wave-tile.
//
// Weights are staged in LDS in a fragment-native layout:
//   element (k, col)  ->  dword  ((k>>2)*64 + col)*4 + (k&3)
// so a B fragment (K-pair for one lane group) is one aligned ds_load_b64, and a
// full wave fragment fetch touches 64 consecutive dwords = all 64 banks once.
// ---------------------------------------------------------------------------------
__global__ void gemm_kernel(const float* __restrict__ F,
                            const float* __restrict__ Wg, const float* __restrict__ bg,
                            const float* __restrict__ a_s, const float* __restrict__ a_d,
                            const float* __restrict__ Wp, const float* __restrict__ bp,
                            float* __restrict__ hg, float* __restrict__ hp,
                            float* __restrict__ ssrc, float* __restrict__ sdst, int n) {
    __shared__ float sW[2 * D * D];      // fragment-native Wg then Wp (32 KB)
    __shared__ float sH[4 * 16 * D];     // per-wave h_gat tile for the a-dots (16 KB)

    const int tid  = threadIdx.x;
    const int wave = tid >> 5;
    const int lane = tid & 31;

    // stage both weight matrices into LDS in fragment-native order
    {
        const float4* g4 = (const float4*)Wg;
        const float4* p4 = (const float4*)Wp;
        for (int i = tid; i < 2 * D * D / 4; i += blockDim.x) {
            const int m    = i >> 10;          // 0 = Wg, 1 = Wp
            const int j    = i & 1023;         // float4 index within matrix
            const int k    = j >> 4;           // row (K)
            const int col0 = (j & 15) * 4;     // first of 4 cols
            const float4 v = m ? p4[j] : g4[j];
            float* dst = sW + m * D * D;
            const int b = ((k >> 2) * D) * 4 + (k & 3);
            dst[b + (col0 + 0) * 4] = v.x;
            dst[b + (col0 + 1) * 4] = v.y;
            dst[b + (col0 + 2) * 4] = v.z;
            dst[b + (col0 + 3) * 4] = v.w;
        }
    }
    __syncthreads();

    const int rowbase = (blockIdx.x * 4 + wave) * 16;
    const int r       = lane & 15;         // row within A half / col within B tile
    const int kh2     = lane >> 4;         // lane group: 0 -> K+0/1, 1 -> K+2/3
    const int ldrow   = min(rowbase + r, n - 1);   // clamped load row (no branches)

    v8f accg[4] = {}, accp[4] = {};

    const float2* F2  = (const float2*)F;
    const v2f*    sW2 = (const v2f*)sW;    // pair view of LDS weights

#pragma unroll
    for (int kk = 0; kk < 16; ++kk) {
        const int kb = kk * 4 + kh2 * 2;   // even -> float2-aligned
        const float2 av = F2[(ldrow * D + kb) >> 1];
        v2f a; a.x = av.x; a.y = av.y;
#pragma unroll
        for (int nt = 0; nt < 4; ++nt) {
            const int col  = nt * 16 + r;
            const int pidx = (kk * D + col) * 2 + kh2;       // v2f index
            const v2f bg_f = sW2[pidx];
            const v2f bp_f = sW2[(D * D / 2) + pidx];
            accg[nt] = __builtin_amdgcn_wmma_f32_16x16x4_f32(
                false, a, false, bg_f, (short)0, accg[nt], false, false);
            accp[nt] = __builtin_amdgcn_wmma_f32_16x16x4_f32(
                false, a, false, bp_f, (short)0, accp[nt], false, false);
        }
    }

    // C/D layout: VGPR rr, lanes 0-15 -> M=rr ; lanes 16-31 -> M=8+rr ; N = lane&15
    const int rhi = kh2 * 8;
#pragma unroll
    for (int nt = 0; nt < 4; ++nt) {
        const int col = nt * 16 + r;
        const float bgv = bg[col];
        const float bpv = bp[col];
#pragma unroll
        for (int rr = 0; rr < 8; ++rr) {
            const int lrow = rr + rhi;
            const int grow = rowbase + lrow;
            const float vg = accg[nt][rr] + bgv;
            const float vp = accp[nt][rr] + bpv;
            sH[wave * (16 * D) + lrow * D + col] = vg;
            if (grow < n) {
                hg[grow * D + col] = vg;
                hp[grow * D + col] = vp;
            }
        }
    }
    __syncthreads();

    // per-row attention dots: lanes 0-15 -> a_src, lanes 16-31 -> a_dst
    {
        const int drow = lane & 15;
        const float* hrow = &sH[wave * (16 * D) + drow * D];
        const float* avec = (lane < 16) ? a_s : a_d;
        float s = 0.0f;
#pragma unroll
        for (int c = 0; c < D; ++c) s = fmaf(hrow[c], avec[c], s);
        const int grow = rowbase + drow;
        if (grow < n) {
            if (lane < 16) ssrc[grow] = s; else sdst[grow] = s;
        }
    }
}

// ---------------------------------------------------------------------------------
// GAT pass 1: per-dst segment max of leaky_relu(s_src[src] + s_dst[dst])
// ---------------------------------------------------------------------------------
__global__ void gat_max_kernel(const int* __restrict__ idx, const float* __restrict__ ssrc,
                               const float* __restrict__ sdst, unsigned* __restrict__ m,
                               int ne) {
    int i = blockIdx.x * blockDim.x + threadIdx.x;
    if (i >= ne) return;
    const int s = idx[i];
    const int d = idx[ne + i];
    const float e = lrelu(ssrc[s] + sdst[d]);
    atomicMax(&m[d], fenc(e));
}

// ---------------------------------------------------------------------------------
// GAT pass 2: ex = exp(e - m[dst]) ; denom[dst] += ex ; store ex
// ---------------------------------------------------------------------------------
__global__ void gat_exp_kernel(const int* __restrict__ idx, const float* __restrict__ ssrc,
                               const float* __restrict__ sdst, const unsigned* __restrict__ m,
                               float* __restrict__ denom, float* __restrict__ ex, int ne) {
    int i = blockIdx.x * blockDim.x + threadIdx.x;
    if (i >= ne) return;
    const int s = idx[i];
    const int d = idx[ne + i];
    const float e = lrelu(ssrc[s] + sdst[d]);
    const float x = expf(e - fdec(m[d]));
    ex[i] = x;
    atomicAdd(&denom[d], x);
}

// ---------------------------------------------------------------------------------
// GAT pass 3: out[dst] += (ex/denom[dst]) * h[src]   -- 16 lanes per edge, float4 rows
// ---------------------------------------------------------------------------------
__global__ void gat_scatter_kernel(const int* __restrict__ idx, const float* __restrict__ h,
                                   const float* __restrict__ ex, const float* __restrict__ denom,
                                   float* __restrict__ out, int ne) {
    int t = blockIdx.x * blockDim.x + threadIdx.x;
    int e = t >> 4;
    int lane = t & 15;
    if (e >= ne) return;
    const int s = idx[e];
    const int d = idx[ne + e];
    const float alpha = ex[e] / (denom[d] + 1e-16f);
    const float4 hv = ((const float4*)h)[s * (D / 4) + lane];
    float* op = out + d * D + lane * 4;
    atomicAdd(op + 0, alpha * hv.x);
    atomicAdd(op + 1, alpha * hv.y);
    atomicAdd(op + 2, alpha * hv.z);
    atomicAdd(op + 3, alpha * hv.w);
}

// ---------------------------------------------------------------------------------
// h_p: out[row] += p_values[e] * hp[col]
// ---------------------------------------------------------------------------------
__global__ void p_scatter_kernel(const int* __restrict__ idx, const float* __restrict__ pv,
                                 const float* __restrict__ hp, float* __restrict__ out, int ne) {
    int t = blockIdx.x * blockDim.x + threadIdx.x;
    int e = t >> 4;
    int lane = t & 15;
    if (e >= ne) return;
    const int row = idx[e];        // scatter target
    const int col = idx[ne + e];   // gather source
    const float w = pv[e];
    const float4 hv = ((const float4*)hp)[col * (D / 4) + lane];
    float* op = out + row * D + lane * 4;
    atomicAdd(op + 0, w * hv.x);
    atomicAdd(op + 1, w * hv.y);
    atomicAdd(op + 2, w * hv.z);
    atomicAdd(op + 3, w * hv.w);
}

// ---------------------------------------------------------------------------------
extern "C" void kernel_launch(void* const* d_in, const int* in_sizes, int n_in,
                              void* d_out, int out_size, void* d_ws, size_t ws_size,
                              hipStream_t stream) {
    const float* features = (const float*)d_in[0];
    const float* W_gat    = (const float*)d_in[1];
    const float* b_gat    = (const float*)d_in[2];
    const float* a_src    = (const float*)d_in[3];
    const float* a_dst    = (const float*)d_in[4];
    const float* W_p      = (const float*)d_in[5];
    const float* b_p      = (const float*)d_in[6];
    const float* p_values = (const float*)d_in[7];
    const int*   lu_idx   = (const int*)d_in[8];
    const int*   ld_idx   = (const int*)d_in[9];
    const int*   p_idx    = (const int*)d_in[10];

    const int n  = in_sizes[0] / D;     // 50000
    const int ne = in_sizes[7];         // 1600000

    // workspace layout (floats unless noted)
    char* ws = (char*)d_ws;
    float*    hg    = (float*)ws;                        ws += (size_t)n * D * 4;
    float*    hp    = (float*)ws;                        ws += (size_t)n * D * 4;
    float*    ssrc  = (float*)ws;                        ws += (size_t)n * 4;
    float*    sdst  = (float*)ws;                        ws += (size_t)n * 4;
    float*    den_u = (float*)ws;                        ws += (size_t)n * 4;
    float*    den_d = (float*)ws;                        ws += (size_t)n * 4;
    unsigned* m_u   = (unsigned*)ws;                     ws += (size_t)n * 4;
    unsigned* m_d   = (unsigned*)ws;                     ws += (size_t)n * 4;
    float*    ex_u  = (float*)ws;                        ws += (size_t)ne * 4;
    float*    ex_d  = (float*)ws;

    float* out = (float*)d_out;
    const int total_out = n * D;

    // 1) init
    {
        int threads = 256;
        int blocks = (total_out + threads - 1) / threads;
        init_kernel<<<blocks, threads, 0, stream>>>(out, den_u, den_d, m_u, m_d, n, total_out);
    }
    // 2) dense GEMMs + attention dots (WMMA)
    {
        int threads = 128;                        // 4 waves x 16 rows = 64 rows/block
        int blocks = (n + 63) / 64;
        gemm_kernel<<<blocks, threads, 0, stream>>>(features, W_gat, b_gat, a_src, a_dst,
                                                    W_p, b_p, hg, hp, ssrc, sdst, n);
    }
    // 3) GAT softmax passes for both edge lists
    {
        int threads = 256;
        int blocks = (ne + threads - 1) / threads;
        gat_max_kernel<<<blocks, threads, 0, stream>>>(lu_idx, ssrc, sdst, m_u, ne);
        gat_max_kernel<<<blocks, threads, 0, stream>>>(ld_idx, ssrc, sdst, m_d, ne);
        gat_exp_kernel<<<blocks, threads, 0, stream>>>(lu_idx, ssrc, sdst, m_u, den_u, ex_u, ne);
        gat_exp_kernel<<<blocks, threads, 0, stream>>>(ld_idx, ssrc, sdst, m_d, den_d, ex_d, ne);
    }
    // 4) weighted message scatter (h_u + h_d) and sparse-mm term (h_p)
    {
        int threads = 256;                        // 16 edges/block, 16 lanes/edge
        long long tot = (long long)ne * 16;
        int blocks = (int)((tot + threads - 1) / threads);
        gat_scatter_kernel<<<blocks, threads, 0, stream>>>(lu_idx, hg, ex_u, den_u, out, ne);
        gat_scatter_kernel<<<blocks, threads, 0, stream>>>(ld_idx, hg, ex_d, den_d, out, ne);
        p_scatter_kernel<<<blocks, threads, 0, stream>>>(p_idx, p_values, hp, out, ne);
    }
}